// Agent2Agent_emb_30537217474895
// MI455X (gfx1250) — compile-verified
//
#include <hip/hip_runtime.h>

typedef float v2f __attribute__((ext_vector_type(2)));
typedef float v8f __attribute__((ext_vector_type(8)));

#define BB 8
#define N1 256
#define N2 256
#define DD 7
#define EE 8
#define HH 128
#define EPITCH 9   // padded LDS row pitch (dwords) -> conflict-free A-operand loads

__global__ __launch_bounds__(256) void a2a_emb_kernel(
    const float* __restrict__ agent1, const float* __restrict__ agent2,
    const float* __restrict__ W, const float* __restrict__ bias,
    float* __restrict__ out)
{
  __shared__ float sEdge[N2 * EPITCH];  // 9216 B
  __shared__ float sW[EE * HH];         // 4096 B, transposed: sW[e*HH + h] = W[h,e]
  __shared__ float sBias[HH];           // 512 B
  __shared__ float sA1[DD];

  const int b    = blockIdx.y;
  const int mrow = blockIdx.x;
  const int tid  = threadIdx.x;

  // ---- stage small operands ----
  if (tid < DD) sA1[tid]  = agent1[((size_t)b * N1 + mrow) * DD + tid];
  if (tid < HH) sBias[tid] = bias[tid];
  for (int i = tid; i < HH * EE; i += 256) {   // i = h*EE + e  (coalesced read of W)
    const int h = i / EE, e = i % EE;
    sW[e * HH + h] = W[i];
  }
  __syncthreads();

  // ---- phase 1: each thread computes edge row n = tid into LDS ----
  {
    float x1[DD];
    #pragma unroll
    for (int k = 0; k < DD; ++k) x1[k] = sA1[k];
    const float* a2p = agent2 + ((size_t)b * N2 + tid) * DD;
    float x2[DD];
    #pragma unroll
    for (int k = 0; k < DD; ++k) x2[k] = a2p[k];

    bool z1 = true, z2 = true;
    #pragma unroll
    for (int k = 0; k < DD; ++k) { z1 = z1 && (x1[k] == 0.0f); z2 = z2 && (x2[k] == 0.0f); }
    const float msk = (z1 || z2) ? 0.0f : 1.0f;   // m = f1 * f2

    const float dxp = x2[0] - x1[0];
    const float dyp = x2[1] - x1[1];
    const float s1 = x1[3], c1 = x1[4];
    const float s2 = x2[3], c2 = x2[4];
    const float dxr = ( c1 * dxp + s1 * dyp) * msk;
    const float dyr = (-s1 * dxp + c1 * dyp) * msk;
    const float dsv = ( s2 * c1 - c2 * s1) * msk;
    const float dcv = ( c2 * c1 + s2 * s1) * msk;
    const float v2  = x2[2];

    float* er = &sEdge[tid * EPITCH];
    er[0] = dxr * 0.1f;     // dxr / XY_SCALE
    er[1] = dyr * 0.1f;     // dyr / XY_SCALE
    er[2] = dsv;
    er[3] = dcv;
    er[4] = v2 * dsv;
    er[5] = v2 * dcv;
    er[6] = x2[5];          // extras: broadcast, unmasked
    er[7] = x2[6];
  }
  __syncthreads();

  // ---- phase 2: WMMA f32 16x16x4, E=8 -> 2 k-steps per 16x16 tile ----
  const int wave  = tid >> 5;
  const int lane  = tid & 31;
  const int lhalf = lane >> 4;        // 0 or 1
  const int l16   = lane & 15;
  const int khalf = lhalf << 1;       // K offset contributed by lane half
  const int hcol  = wave * 16 + l16;  // wave w owns h-tile w

  // B operand: VGPR i, lane half j -> K = kbase + i + 2j, N = lane&15 (loop-invariant)
  const v2f bm0 = { sW[(0 + khalf) * HH + hcol], sW[(1 + khalf) * HH + hcol] };  // e = 0..3
  const v2f bm1 = { sW[(4 + khalf) * HH + hcol], sW[(5 + khalf) * HH + hcol] };  // e = 4..7

  const float cinit = sBias[hcol];
  const v8f cbase = { cinit, cinit, cinit, cinit, cinit, cinit, cinit, cinit };

  float* outBase = out + ((size_t)b * N1 + mrow) * N2 * HH;

  #pragma unroll 4
  for (int nt = 0; nt < 16; ++nt) {
    const int nrow = nt * 16 + l16;               // A: M = lane&15
    const float* er = &sEdge[nrow * EPITCH];
    const v2f a0 = { er[khalf],     er[khalf + 1]     };  // K = 0..3
    const v2f a1 = { er[4 + khalf], er[5 + khalf]     };  // K = 4..7

    v8f c = cbase;
    c = __builtin_amdgcn_wmma_f32_16x16x4_f32(false, a0, false, bm0, (short)0, c, false, false);
    c = __builtin_amdgcn_wmma_f32_16x16x4_f32(false, a1, false, bm1, (short)0, c, false, false);

    // C: VGPR j holds (n = nt*16 + j + 8*lhalf, h = hcol)
    float* op = outBase + ((size_t)(nt * 16 + 8 * lhalf)) * HH + hcol;
    #pragma unroll
    for (int j = 0; j < 8; ++j) op[(size_t)j * HH] = c[j];
  }
}

extern "C" void kernel_launch(void* const* d_in, const int* in_sizes, int n_in,
                              void* d_out, int out_size, void* d_ws, size_t ws_size,
                              hipStream_t stream) {
  const float* agent1 = (const float*)d_in[0];
  const float* agent2 = (const float*)d_in[1];
  const float* W      = (const float*)d_in[2];
  const float* bias   = (const float*)d_in[3];
  float* out = (float*)d_out;

  dim3 grid(N1, BB);   // one workgroup per (b, m): 2048 workgroups, 8 waves each
  a2a_emb_kernel<<<grid, dim3(256), 0, stream>>>(agent1, agent2, W, bias, out);
}